// StateSpaceModel_52166672777475
// MI455X (gfx1250) — compile-verified
//
#include <hip/hip_runtime.h>
#include <math.h>

typedef __attribute__((ext_vector_type(2))) float v2f;
typedef __attribute__((ext_vector_type(8))) float v8f;

#define T_STEPS 128
#define BATCH   64
#define ADIM    32
#define ZDIM    64
#define KMIX    16
#define HIDDEN  128

// ---------- fp32 WMMA 16x16x4 helpers ----------
__device__ __forceinline__ v8f wmma4(v2f a, v2f b, v8f c) {
    return __builtin_amdgcn_wmma_f32_16x16x4_f32(
        false, a, false, b, (short)0, c, false, false);
}

// A fragment (16x4): rows rowBase..+15, k kBase..+3 from row-major src (leading dim ld)
__device__ __forceinline__ v2f ldA(const float* src, int ld, int rowBase, int kBase,
                                   int lm, int half) {
    const float* p = src + (rowBase + lm) * ld + kBase + 2 * half;
    v2f a; a.x = p[0]; a.y = p[1]; return a;
}
// A fragment where A[m][k] = srcT[k][m]   (srcT row-major K x M, leading dim ld)
__device__ __forceinline__ v2f ldAT(const float* srcT, int ld, int rowBase, int kBase,
                                    int lm, int half) {
    v2f a;
    a.x = srcT[(kBase + 2 * half    ) * ld + rowBase + lm];
    a.y = srcT[(kBase + 2 * half + 1) * ld + rowBase + lm];
    return a;
}
// B fragment (4x16): k rows kBase..+3, cols colBase..+15 from row-major K x N src
__device__ __forceinline__ v2f ldB(const float* src, int ld, int kBase, int colBase,
                                   int lm, int half) {
    v2f b;
    b.x = src[(kBase + 2 * half    ) * ld + colBase + lm];
    b.y = src[(kBase + 2 * half + 1) * ld + colBase + lm];
    return b;
}
// B fragment where B[k][n] = srcT[n][k]  (srcT row-major N x K)
__device__ __forceinline__ v2f ldBT(const float* srcT, int ld, int kBase, int colBase,
                                    int lm, int half) {
    const float* p = srcT + (colBase + lm) * ld + kBase + 2 * half;
    v2f b; b.x = p[0]; b.y = p[1]; return b;
}

__device__ __forceinline__ float sigmoidf_(float x) {
    return 1.0f / (1.0f + __expf(-x));
}

// ============================================================================
// Kernel 1: fused 2-layer LSTM. Batch split over 4 blocks (16 rows each);
// layer-1 consumes layer-0's h_t from LDS in the same timestep.
// Per wave: one 16-col tile of hidden, all 4 gates, both layers.
// ============================================================================
__global__ void __launch_bounds__(256)
lstm2_kernel(const float* __restrict__ x,
             const float* __restrict__ Wih0, const float* __restrict__ Whh0,
             const float* __restrict__ bih0, const float* __restrict__ bhh0,
             const float* __restrict__ Wih1, const float* __restrict__ Whh1,
             const float* __restrict__ bih1, const float* __restrict__ bhh1,
             float* __restrict__ h1out) {
    __shared__ float xs [16 * ADIM];     // layer-0 input staging
    __shared__ float c0 [16 * HIDDEN];
    __shared__ float h0A[16 * HIDDEN];
    __shared__ float h0B[16 * HIDDEN];
    __shared__ float c1 [16 * HIDDEN];
    __shared__ float h1A[16 * HIDDEN];
    __shared__ float h1B[16 * HIDDEN];

    const int rb   = blockIdx.x * 16;    // global batch-row base
    const int tid  = threadIdx.x;
    const int wave = tid >> 5;           // 0..7 -> hidden col tile
    const int lane = tid & 31;
    const int lm   = lane & 15;
    const int half = lane >> 4;
    const int colBase = wave * 16;

    for (int i = tid; i < 16 * HIDDEN; i += 256) {
        c0[i] = 0.0f; h0A[i] = 0.0f; c1[i] = 0.0f; h1A[i] = 0.0f;
    }
    __syncthreads();

    for (int t = 0; t < T_STEPS; ++t) {
        const float* h0cur = (t & 1) ? h0B : h0A;
        float*       h0nxt = (t & 1) ? h0A : h0B;
        const float* h1cur = (t & 1) ? h1B : h1A;
        float*       h1nxt = (t & 1) ? h1A : h1B;

        for (int i = tid; i < 16 * ADIM; i += 256)
            xs[i] = x[(t * BATCH + rb) * ADIM + i];
        __syncthreads();

        // ---------------- layer 0 (in_dim = 32) ----------------
        {
            v8f acc[4];
            #pragma unroll
            for (int g = 0; g < 4; ++g) {
                const int col = g * HIDDEN + colBase + lm;
                const float bv = bih0[col] + bhh0[col];
                #pragma unroll
                for (int r = 0; r < 8; ++r) acc[g][r] = bv;
            }
            for (int k = 0; k < ADIM; k += 4) {
                v2f a = ldA(xs, ADIM, 0, k, lm, half);
                #pragma unroll
                for (int g = 0; g < 4; ++g) {
                    v2f b = ldBT(Wih0, ADIM, k, g * HIDDEN + colBase, lm, half);
                    acc[g] = wmma4(a, b, acc[g]);
                }
            }
            for (int k = 0; k < HIDDEN; k += 4) {
                v2f a = ldA(h0cur, HIDDEN, 0, k, lm, half);
                #pragma unroll
                for (int g = 0; g < 4; ++g) {
                    v2f b = ldBT(Whh0, HIDDEN, k, g * HIDDEN + colBase, lm, half);
                    acc[g] = wmma4(a, b, acc[g]);
                }
            }
            #pragma unroll
            for (int r = 0; r < 8; ++r) {
                const int row = half * 8 + r;
                const int col = colBase + lm;
                const float iv = sigmoidf_(acc[0][r]);
                const float fv = sigmoidf_(acc[1][r]);
                const float gv = tanhf(acc[2][r]);
                const float ov = sigmoidf_(acc[3][r]);
                const float cv = fv * c0[row * HIDDEN + col] + iv * gv;
                c0[row * HIDDEN + col] = cv;
                h0nxt[row * HIDDEN + col] = ov * tanhf(cv);
            }
        }
        __syncthreads();   // h0_t complete -> layer-1 input ready

        // ---------------- layer 1 (in = h0_t, 128) ----------------
        {
            v8f acc[4];
            #pragma unroll
            for (int g = 0; g < 4; ++g) {
                const int col = g * HIDDEN + colBase + lm;
                const float bv = bih1[col] + bhh1[col];
                #pragma unroll
                for (int r = 0; r < 8; ++r) acc[g][r] = bv;
            }
            for (int k = 0; k < HIDDEN; k += 4) {
                v2f a = ldA(h0nxt, HIDDEN, 0, k, lm, half);
                #pragma unroll
                for (int g = 0; g < 4; ++g) {
                    v2f b = ldBT(Wih1, HIDDEN, k, g * HIDDEN + colBase, lm, half);
                    acc[g] = wmma4(a, b, acc[g]);
                }
            }
            for (int k = 0; k < HIDDEN; k += 4) {
                v2f a = ldA(h1cur, HIDDEN, 0, k, lm, half);
                #pragma unroll
                for (int g = 0; g < 4; ++g) {
                    v2f b = ldBT(Whh1, HIDDEN, k, g * HIDDEN + colBase, lm, half);
                    acc[g] = wmma4(a, b, acc[g]);
                }
            }
            #pragma unroll
            for (int r = 0; r < 8; ++r) {
                const int row = half * 8 + r;
                const int col = colBase + lm;
                const float iv = sigmoidf_(acc[0][r]);
                const float fv = sigmoidf_(acc[1][r]);
                const float gv = tanhf(acc[2][r]);
                const float ov = sigmoidf_(acc[3][r]);
                const float cv = fv * c1[row * HIDDEN + col] + iv * gv;
                c1[row * HIDDEN + col] = cv;
                const float hv = ov * tanhf(cv);
                h1nxt[row * HIDDEN + col] = hv;
                h1out[((long)t * BATCH + rb + row) * HIDDEN + col] = hv;
            }
        }
        __syncthreads();
    }
}

// ============================================================================
// Kernel 2: logits = h1 @ lin_W^T + lin_b ; softmax over K=16. One row/thread.
// ============================================================================
__global__ void __launch_bounds__(256)
head_kernel(const float* __restrict__ h1, const float* __restrict__ linW,
            const float* __restrict__ linb, float* __restrict__ wmix) {
    const int r = blockIdx.x * 256 + threadIdx.x;   // 0..8191
    const float* hv = h1 + (long)r * HIDDEN;
    float lg[KMIX];
    float mx = -1e30f;
    for (int k = 0; k < KMIX; ++k) {
        float s = linb[k];
        const float* wr = linW + k * HIDDEN;
        for (int d = 0; d < HIDDEN; ++d) s += hv[d] * wr[d];
        lg[k] = s; mx = fmaxf(mx, s);
    }
    float den = 0.0f;
    for (int k = 0; k < KMIX; ++k) { lg[k] = __expf(lg[k] - mx); den += lg[k]; }
    const float inv = 1.0f / den;
    for (int k = 0; k < KMIX; ++k) wmix[(long)r * KMIX + k] = lg[k] * inv;
}

// ============================================================================
// Kernel 3: mat_As = w @ A_K (flat [16,4096]), mat_Cs = w @ C_K ([16,2048]).
// One 16x16 output tile per wave, K=16 => 4 WMMAs.
// ============================================================================
__global__ void __launch_bounds__(256)
mix_kernel(const float* __restrict__ w, const float* __restrict__ AK,
           const float* __restrict__ CK, float* __restrict__ As,
           float* __restrict__ Cs) {
    const int wave = threadIdx.x >> 5;
    const int lane = threadIdx.x & 31;
    const int lm = lane & 15, half = lane >> 4;

    const int ATILES = 512 * 256;   // M-tiles x N-tiles for mat_As
    int gt = blockIdx.x * 8 + wave;

    const float* B; float* D; int mt, nt, N;
    if (gt < ATILES) { mt = gt >> 8; nt = gt & 255; B = AK; D = As; N = ZDIM * ZDIM; }
    else { int g = gt - ATILES; mt = g >> 7; nt = g & 127; B = CK; D = Cs; N = ADIM * ZDIM; }

    v8f acc = {};
    #pragma unroll
    for (int k = 0; k < KMIX; k += 4) {
        v2f a = ldA(w, KMIX, mt * 16, k, lm, half);
        v2f b = ldB(B, N, k, nt * 16, lm, half);
        acc = wmma4(a, b, acc);
    }
    #pragma unroll
    for (int r = 0; r < 8; ++r) {
        const int row = mt * 16 + half * 8 + r;
        const int col = nt * 16 + lm;
        D[(long)row * N + col] = acc[r];
    }
}

// ============================================================================
// Kernel 4: Q = Q_L Q_L^T + 1e-3 I ; R = R_L R_L^T + 1e-3 I
// ============================================================================
__global__ void __launch_bounds__(256)
qr_kernel(const float* __restrict__ QL, const float* __restrict__ RL,
          float* __restrict__ Qm, float* __restrict__ Rm) {
    const int tid = blockIdx.x * 256 + threadIdx.x;
    if (tid < ZDIM * ZDIM) {
        const int i = tid >> 6, j = tid & 63;
        float s = (i == j) ? 1e-3f : 0.0f;
        for (int k = 0; k < ZDIM; ++k) s += QL[i * ZDIM + k] * QL[j * ZDIM + k];
        Qm[tid] = s;
    } else if (tid < ZDIM * ZDIM + ADIM * ADIM) {
        const int e = tid - ZDIM * ZDIM;
        const int i = e >> 5, j = e & 31;
        float s = (i == j) ? 1e-3f : 0.0f;
        for (int k = 0; k < ADIM; ++k) s += RL[i * ADIM + k] * RL[j * ADIM + k];
        Rm[e] = s;
    }
}

// ============================================================================
// Kernel 5: Kalman scan. One block (8 waves) per batch element b.
// Covariance P (64x64) lives in LDS across all 128 steps.
// Prefetches step t+1's A/C tiles (global_prefetch_b8) behind step t compute.
// ============================================================================
__global__ void __launch_bounds__(256)
kalman_kernel(const float* __restrict__ as_, const float* __restrict__ Qm,
              const float* __restrict__ Rm,
              const float* __restrict__ As, const float* __restrict__ Cs,
              float* __restrict__ means, float* __restrict__ covs,
              float* __restrict__ nmeans, float* __restrict__ ncovs) {
    __shared__ float P [ZDIM * ZDIM];   // predicted covariance
    __shared__ float At[ZDIM * ZDIM];
    __shared__ float Ct[ADIM * ZDIM];
    __shared__ float Mm[ADIM * ZDIM];   // M = C P
    __shared__ float SI[ADIM * 2 * ADIM]; // [S | I] -> [I | S^-1]
    __shared__ float Yb[ADIM * ZDIM];   // Y = S^-1 M
    __shared__ float CT[ZDIM * ZDIM];   // cov_t (sym) / raw cov_next
    __shared__ float TM[ZDIM * ZDIM];   // scratch product
    __shared__ float meanp[ZDIM], meant[ZDIM], meann[ZDIM];
    __shared__ float innov[ADIM], uvec[ADIM], fac[ADIM];
    __shared__ float spiv;

    const int b = blockIdx.x;
    const int tid  = threadIdx.x;
    const int wave = tid >> 5;
    const int lane = tid & 31;
    const int lm = lane & 15, half = lane >> 4;

    for (int i = tid; i < ZDIM * ZDIM; i += 256)
        P[i] = ((i >> 6) == (i & 63)) ? 1.0f : 0.0f;
    if (tid < ZDIM) meanp[tid] = 0.0f;
    __syncthreads();

    for (int t = 0; t < T_STEPS; ++t) {
        const long tb = (long)t * BATCH + b;
        const float* Asrc = As + tb * (ZDIM * ZDIM);
        const float* Csrc = Cs + tb * (ADIM * ZDIM);
        for (int i = tid; i < ZDIM * ZDIM; i += 256) At[i] = Asrc[i];
        for (int i = tid; i < ADIM * ZDIM; i += 256) Ct[i] = Csrc[i];

        // prefetch t+1's A/C tiles (128B cachelines) while step t computes
        if (t + 1 < T_STEPS) {
            const float* An = Asrc + (long)BATCH * (ZDIM * ZDIM);
            const float* Cn = Csrc + (long)BATCH * (ADIM * ZDIM);
            if (tid < 192) {
                if (tid < 128) __builtin_prefetch(An + tid * 32, 0, 3);
                else           __builtin_prefetch(Cn + (tid - 128) * 32, 0, 3);
            }
        }
        __syncthreads();

        // ---- M = Ct @ P : 32x64, K=64 -> 8 tiles, one per wave
        {
            const int mt = wave >> 2, nt = wave & 3;
            v8f acc = {};
            for (int k = 0; k < ZDIM; k += 4) {
                v2f a = ldA(Ct, ZDIM, mt * 16, k, lm, half);
                v2f bb = ldB(P, ZDIM, k, nt * 16, lm, half);
                acc = wmma4(a, bb, acc);
            }
            #pragma unroll
            for (int r = 0; r < 8; ++r)
                Mm[(mt * 16 + half * 8 + r) * ZDIM + nt * 16 + lm] = acc[r];
        }
        __syncthreads();

        // ---- S = M @ Ct^T + R into SI[:,0:32]; identity into SI[:,32:64]
        if (wave < 4) {
            const int mt = wave >> 1, nt = wave & 1;
            v8f acc = {};
            for (int k = 0; k < ZDIM; k += 4) {
                v2f a = ldA(Mm, ZDIM, mt * 16, k, lm, half);
                v2f bb = ldBT(Ct, ZDIM, k, nt * 16, lm, half);
                acc = wmma4(a, bb, acc);
            }
            #pragma unroll
            for (int r = 0; r < 8; ++r) {
                const int row = mt * 16 + half * 8 + r, col = nt * 16 + lm;
                SI[row * 64 + col] = acc[r] + Rm[row * ADIM + col];
            }
        } else {
            const int id = tid - 128;
            for (int e = id; e < ADIM * ADIM; e += 128) {
                const int i = e >> 5, j = e & 31;
                SI[i * 64 + 32 + j] = (i == j) ? 1.0f : 0.0f;
            }
        }
        __syncthreads();

        // ---- Gauss-Jordan: [S|I] -> [I|S^-1]  (S is SPD, no pivoting)
        for (int k = 0; k < ADIM; ++k) {
            if (tid == 0) spiv = 1.0f / SI[k * 64 + k];
            if (tid < ADIM) fac[tid] = SI[tid * 64 + k];
            __syncthreads();
            if (tid < 64) SI[k * 64 + tid] *= spiv;
            __syncthreads();
            for (int e = tid; e < ADIM * 64; e += 256) {
                const int i = e >> 6, j = e & 63;
                if (i != k) SI[i * 64 + j] -= fac[i] * SI[k * 64 + j];
            }
            __syncthreads();
        }

        // ---- Y = S^-1 @ M : 32x64, K=32 -> 8 tiles
        {
            const int mt = wave >> 2, nt = wave & 3;
            v8f acc = {};
            for (int k = 0; k < ADIM; k += 4) {
                v2f a = ldA(SI + 32, 64, mt * 16, k, lm, half);  // S^-1 block
                v2f bb = ldB(Mm, ZDIM, k, nt * 16, lm, half);
                acc = wmma4(a, bb, acc);
            }
            #pragma unroll
            for (int r = 0; r < 8; ++r)
                Yb[(mt * 16 + half * 8 + r) * ZDIM + nt * 16 + lm] = acc[r];
        }
        __syncthreads();

        // ---- innovation, u = S^-1 innov, mean_t = mean_p + M^T u
        if (tid < ADIM) {
            float s = as_[tb * ADIM + tid];
            for (int z = 0; z < ZDIM; ++z) s -= Ct[tid * ZDIM + z] * meanp[z];
            innov[tid] = s;
        }
        __syncthreads();
        if (tid < ADIM) {
            float s = 0.0f;
            for (int j = 0; j < ADIM; ++j) s += SI[tid * 64 + 32 + j] * innov[j];
            uvec[tid] = s;
        }
        __syncthreads();
        if (tid < ZDIM) {
            float s = meanp[tid];
            for (int i = 0; i < ADIM; ++i) s += Mm[i * ZDIM + tid] * uvec[i];
            meant[tid] = s;
            means[tb * ZDIM + tid] = s;
        }
        __syncthreads();

        // ---- raw cov_t = P - M^T @ Y : 64x64, K=32 -> 16 tiles (2 per wave)
        for (int tt = wave; tt < 16; tt += 8) {
            const int mt = tt >> 2, nt = tt & 3;
            v8f acc = {};
            for (int k = 0; k < ADIM; k += 4) {
                v2f a = ldAT(Mm, ZDIM, mt * 16, k, lm, half);   // A[m][k] = M[k][m]
                v2f bb = ldB(Yb, ZDIM, k, nt * 16, lm, half);
                acc = wmma4(a, bb, acc);
            }
            #pragma unroll
            for (int r = 0; r < 8; ++r) {
                const int row = mt * 16 + half * 8 + r, col = nt * 16 + lm;
                TM[row * ZDIM + col] = P[row * ZDIM + col] - acc[r];
            }
        }
        __syncthreads();
        // symmetrize -> CT, write covs output
        {
            float* outp = covs + tb * (ZDIM * ZDIM);
            for (int e = tid; e < ZDIM * ZDIM; e += 256) {
                const int i = e >> 6, j = e & 63;
                const float v = 0.5f * (TM[i * ZDIM + j] + TM[j * ZDIM + i]);
                CT[e] = v;
                outp[e] = v;
            }
        }
        __syncthreads();

        // ---- TM = At @ cov_t : 64x64, K=64
        for (int tt = wave; tt < 16; tt += 8) {
            const int mt = tt >> 2, nt = tt & 3;
            v8f acc = {};
            for (int k = 0; k < ZDIM; k += 4) {
                v2f a = ldA(At, ZDIM, mt * 16, k, lm, half);
                v2f bb = ldB(CT, ZDIM, k, nt * 16, lm, half);
                acc = wmma4(a, bb, acc);
            }
            #pragma unroll
            for (int r = 0; r < 8; ++r)
                TM[(mt * 16 + half * 8 + r) * ZDIM + nt * 16 + lm] = acc[r];
        }
        __syncthreads();

        // ---- raw cov_next = TM @ At^T + Q  (overwrite CT)
        for (int tt = wave; tt < 16; tt += 8) {
            const int mt = tt >> 2, nt = tt & 3;
            v8f acc = {};
            for (int k = 0; k < ZDIM; k += 4) {
                v2f a = ldA(TM, ZDIM, mt * 16, k, lm, half);
                v2f bb = ldBT(At, ZDIM, k, nt * 16, lm, half);
                acc = wmma4(a, bb, acc);
            }
            #pragma unroll
            for (int r = 0; r < 8; ++r) {
                const int row = mt * 16 + half * 8 + r, col = nt * 16 + lm;
                CT[row * ZDIM + col] = acc[r] + Qm[row * ZDIM + col];
            }
        }
        __syncthreads();

        // ---- mean_next = At @ mean_t ; symmetrize cov_next -> P ; outputs
        if (tid < ZDIM) {
            float s = 0.0f;
            for (int z = 0; z < ZDIM; ++z) s += At[tid * ZDIM + z] * meant[z];
            meann[tid] = s;
            nmeans[tb * ZDIM + tid] = s;
        }
        {
            float* outp = ncovs + tb * (ZDIM * ZDIM);
            for (int e = tid; e < ZDIM * ZDIM; e += 256) {
                const int i = e >> 6, j = e & 63;
                const float v = 0.5f * (CT[i * ZDIM + j] + CT[j * ZDIM + i]);
                P[e] = v;
                outp[e] = v;
            }
        }
        __syncthreads();
        if (tid < ZDIM) meanp[tid] = meann[tid];
        __syncthreads();
    }
}

// ============================================================================
extern "C" void kernel_launch(void* const* d_in, const int* in_sizes, int n_in,
                              void* d_out, int out_size, void* d_ws, size_t ws_size,
                              hipStream_t stream) {
    (void)in_sizes; (void)n_in; (void)out_size; (void)ws_size;

    const float* as_  = (const float*)d_in[0];
    const float* AK   = (const float*)d_in[1];   // [16,64,64]
    const float* CK   = (const float*)d_in[2];   // [16,32,64]
    const float* QL   = (const float*)d_in[3];
    const float* RL   = (const float*)d_in[4];
    const float* linW = (const float*)d_in[5];
    const float* linb = (const float*)d_in[6];
    const float* Wih0 = (const float*)d_in[7];
    const float* Whh0 = (const float*)d_in[8];
    const float* bih0 = (const float*)d_in[9];
    const float* bhh0 = (const float*)d_in[10];
    const float* Wih1 = (const float*)d_in[11];
    const float* Whh1 = (const float*)d_in[12];
    const float* bih1 = (const float*)d_in[13];
    const float* bhh1 = (const float*)d_in[14];

    // Output layout (flat concat, reference return order)
    float* out    = (float*)d_out;
    float* means  = out;                               // T*B*64      = 524288
    float* covs   = means  + 524288;                   // T*B*64*64   = 33554432
    float* nmeans = covs   + 33554432;                 // 524288
    float* ncovs  = nmeans + 524288;                   // 33554432
    float* mAs    = ncovs  + 33554432;                 // 33554432
    float* mCs    = mAs    + 33554432;                 // T*B*32*64   = 16777216

    // Workspace
    float* ws   = (float*)d_ws;
    float* h1   = ws;                  // T*B*128 = 1048576
    float* wmix = h1   + 1048576;      // T*B*16  = 131072
    float* Qm   = wmix + 131072;       // 4096
    float* Rm   = Qm   + 4096;         // 1024

    lstm2_kernel<<<4, 256, 0, stream>>>(as_, Wih0, Whh0, bih0, bhh0,
                                        Wih1, Whh1, bih1, bhh1, h1);
    head_kernel<<<32, 256, 0, stream>>>(h1, linW, linb, wmix);
    mix_kernel<<<24576, 256, 0, stream>>>(wmix, AK, CK, mAs, mCs);  // (512*256+512*128)/8 blocks
    qr_kernel<<<20, 256, 0, stream>>>(QL, RL, Qm, Rm);
    kalman_kernel<<<64, 256, 0, stream>>>(as_, Qm, Rm, mAs, mCs,
                                          means, covs, nmeans, ncovs);
}